// CrossModalAttention_71957882077419
// MI455X (gfx1250) — compile-verified
//
#include <hip/hip_runtime.h>
#include <hip/hip_bf16.h>

typedef float v2f __attribute__((ext_vector_type(2)));
typedef float v4f __attribute__((ext_vector_type(4)));
typedef float v8f __attribute__((ext_vector_type(8)));

#define N_   8
#define C_   256
#define T1_  16
#define HW_  784
#define T2_  64
#define V_   25
#define REL_ 32
#define KC_  32   // input-channel chunk staged to LDS per pipeline stage
#define NCH_ (C_ / KC_)

#if defined(__gfx1250__) && __has_builtin(__builtin_amdgcn_global_load_async_to_lds_b32)
#define USE_ASYNC_LDS 1
#else
#define USE_ASYNC_LDS 0
#endif

__device__ __forceinline__ v8f zero8() {
  v8f z;
#pragma unroll
  for (int i = 0; i < 8; ++i) z[i] = 0.0f;
  return z;
}

__device__ __forceinline__ v8f wmma_f32(v2f a, v2f b, v8f c) {
#if defined(__gfx1250__)
  // D = A(16x4) * B(4x16) + C(16x16), fp32
  return __builtin_amdgcn_wmma_f32_16x16x4_f32(false, a, false, b,
                                               (short)0, c, false, false);
#else
  (void)a; (void)b;
  return c;
#endif
}

#if USE_ASYNC_LDS
typedef __attribute__((address_space(1))) int gas_int;
typedef __attribute__((address_space(3))) int las_int;

__device__ __forceinline__ void async_cp_b32(const float* g, float* l) {
  __builtin_amdgcn_global_load_async_to_lds_b32((gas_int*)g, (las_int*)l, 0, 0);
}
__device__ __forceinline__ void wait_async0() {
#if __has_builtin(__builtin_amdgcn_s_wait_asynccnt)
  __builtin_amdgcn_s_wait_asynccnt(0);
#else
  asm volatile("s_wait_asynccnt 0x0" ::: "memory");
#endif
}
#endif

// ---------------------------------------------------------------------------
// Kernel 1: img_mean[n,c,t] = mean over H*W  (one block per row of 784)
// ---------------------------------------------------------------------------
__global__ __launch_bounds__(128) void k_mean(const float* __restrict__ image,
                                              float* __restrict__ mean_ws) {
  __shared__ float red[128];
  const int row = blockIdx.x;                     // (n*C + c)*T1 + t
  const float* src = image + (size_t)row * HW_;
  float s = 0.0f;
  for (int i = threadIdx.x; i < HW_; i += 128) s += src[i];
  red[threadIdx.x] = s;
  __syncthreads();
  for (int off = 64; off > 0; off >>= 1) {
    if (threadIdx.x < off) red[threadIdx.x] += red[threadIdx.x + off];
    __syncthreads();
  }
  if (threadIdx.x == 0) mean_ws[row] = red[0] * (1.0f / HW_);
}

// ---------------------------------------------------------------------------
// Kernel 2: build att[n,c,i,j] = (w_att @ tanh(K_i - Q_j) + b_att)/T1 + I
// One block per n, 256 threads (thread == channel c).
// ---------------------------------------------------------------------------
__global__ __launch_bounds__(256) void k_att(
    const float* __restrict__ motion, const float* __restrict__ mean_ws,
    const float* __restrict__ w_mq, const float* __restrict__ b_mq,
    const float* __restrict__ w_iq, const float* __restrict__ b_iq,
    const float* __restrict__ w_att, const float* __restrict__ b_att,
    float* __restrict__ att_ws) {
  __shared__ float sm[9216];                 // 36 KB
  float* smQ      = sm;                      // [REL][T1]   512
  float* smK      = sm + 512;                // [REL][T1]   512
  float* smMean   = sm + 1024;               // [C][T1]    4096
  float* smPool   = sm + 5120;               // [C][T1]    4096
  float* smScores = sm + 1024;               // [REL][T1][T1] 8192 (overlays mean/pool)

  const int n = blockIdx.x;
  const int c = threadIdx.x;

  for (int i = threadIdx.x; i < C_ * T1_; i += 256)
    smMean[i] = mean_ws[n * C_ * T1_ + i];

  {  // motion pool: block-mean over 4 consecutive t2 and all V
    const float* mrow = motion + (size_t)(n * C_ + c) * (T2_ * V_);
    for (int t = 0; t < T1_; ++t) {
      float s = 0.0f;
      for (int u = 0; u < 4 * V_; ++u) s += mrow[t * 4 * V_ + u];
      smPool[c * T1_ + t] = s * (1.0f / (4 * V_));
    }
  }
  __syncthreads();

  for (int p = threadIdx.x; p < REL_ * T1_; p += 256) {
    const int r = p / T1_, t = p % T1_;
    float q = b_iq[r], k = b_mq[r];
    for (int cc = 0; cc < C_; ++cc) {
      q += w_iq[r * C_ + cc] * smMean[cc * T1_ + t];
      k += w_mq[r * C_ + cc] * smPool[cc * T1_ + t];
    }
    smQ[p] = q;
    smK[p] = k;
  }
  __syncthreads();

  for (int s = threadIdx.x; s < REL_ * T1_ * T1_; s += 256) {
    const int r = s >> 8, ij = s & 255, i = ij >> 4, j = ij & 15;
    smScores[s] = tanhf(smK[r * T1_ + i] - smQ[r * T1_ + j]);
  }
  __syncthreads();

  {
    float wr[REL_];
#pragma unroll
    for (int r = 0; r < REL_; ++r) wr[r] = w_att[c * REL_ + r];
    const float bb = b_att[c] * (1.0f / T1_);
    float* dst = att_ws + (size_t)(n * C_ + c) * 256;
    for (int ij = 0; ij < 256; ++ij) {
      float a = 0.0f;
#pragma unroll
      for (int r = 0; r < REL_; ++r) a += wr[r] * smScores[r * 256 + ij];
      a = a * (1.0f / T1_) + bb + (((ij >> 4) == (ij & 15)) ? 1.0f : 0.0f);
      dst[ij] = a;
    }
  }
}

// ---------------------------------------------------------------------------
// Kernel 3 (hot): fused  Vv = w_iv @ image (+b_iv)  then  y = att @ Vv
// grid = (49 v-tiles, 2 c-blocks of 128, 8 n); 256 threads = 8 waves.
// Wave w owns 16 output channels; 16 f32 WMMA acc tiles (one per j).
// LDS B layout packs 4 consecutive k per (j,v): one ds_load_b64 per B frag.
// Async double-buffered staging when the async-to-LDS builtin is available;
// otherwise synchronous staging + global_prefetch of the next chunk.
// ---------------------------------------------------------------------------
__global__ __launch_bounds__(256) void k_main(
    const float* __restrict__ image, const float* __restrict__ w_iv,
    const float* __restrict__ b_iv, const float* __restrict__ att_ws,
    float* __restrict__ out) {
  __shared__ float smB[USE_ASYNC_LDS ? 2 : 1][KC_ * 256];  // [k4][j][v][kk]
  __shared__ float smVV[8 * 256];    // per-wave [16 v][16 j] scratch, 8 KB

  const int vb   = blockIdx.x;            // 0..48  spatial tile (16 of 784)
  const int cb   = blockIdx.y;            // 0..1   block of 128 out-channels
  const int n    = blockIdx.z;            // 0..7
  const int hw0  = vb * 16;
  const int tid  = threadIdx.x;
  const int wave = tid >> 5;
  const int lane = tid & 31;
  const int lc   = lane & 15;             // N / M-row index within fragment
  const int half = lane >> 4;             // lane-half selects K-pair / M+8
  const int c0   = cb * 128 + wave * 16;  // first out-channel of this wave

  v8f acc[16];
#pragma unroll
  for (int j = 0; j < 16; ++j) acc[j] = zero8();

  const float* imgn = image + (size_t)n * C_ * T1_ * HW_;
  const float* wrow = w_iv + (size_t)(c0 + lc) * C_ + 2 * half;  // A frag row

#if USE_ASYNC_LDS
  // prologue: asynchronously stage chunk 0
  for (int q = tid; q < (KC_ / 4) * 256; q += 256) {
    const int v = q & 15, j = (q >> 4) & 15, k4 = q >> 8;
    const float* p = imgn + (size_t)(k4 * 4) * (T1_ * HW_) + j * HW_ + hw0 + v;
    float* d = &smB[0][q * 4];
    async_cp_b32(p, d);
    async_cp_b32(p + T1_ * HW_, d + 1);
    async_cp_b32(p + 2 * T1_ * HW_, d + 2);
    async_cp_b32(p + 3 * T1_ * HW_, d + 3);
  }
  wait_async0();
  __syncthreads();
#endif

  for (int ck = 0; ck < NCH_; ++ck) {
    const int kb = ck * KC_;
#if USE_ASYNC_LDS
    const int cur = ck & 1;
    if (ck + 1 < NCH_) {  // stage next chunk into the other buffer (overlapped)
      for (int q = tid; q < (KC_ / 4) * 256; q += 256) {
        const int v = q & 15, j = (q >> 4) & 15, k4 = q >> 8;
        const float* p =
            imgn + (size_t)(kb + KC_ + k4 * 4) * (T1_ * HW_) + j * HW_ + hw0 + v;
        float* d = &smB[cur ^ 1][q * 4];
        async_cp_b32(p, d);
        async_cp_b32(p + T1_ * HW_, d + 1);
        async_cp_b32(p + 2 * T1_ * HW_, d + 2);
        async_cp_b32(p + 3 * T1_ * HW_, d + 3);
      }
    }
    const float* bbuf = smB[cur];
#else
    __syncthreads();
    // synchronous stage of this chunk (k-packed layout, b128 LDS stores)
    for (int q = tid; q < (KC_ / 4) * 256; q += 256) {
      const int v = q & 15, j = (q >> 4) & 15, k4 = q >> 8;
      const float* p =
          imgn + (size_t)(kb + k4 * 4) * (T1_ * HW_) + j * HW_ + hw0 + v;
      v4f x;
      x.x = p[0];
      x.y = p[T1_ * HW_];
      x.z = p[2 * T1_ * HW_];
      x.w = p[3 * T1_ * HW_];
      *(v4f*)(&smB[0][q * 4]) = x;
    }
    __syncthreads();
    if (kb + KC_ < C_) {  // warm next chunk's cachelines during compute
      for (int pr = tid; pr < KC_ * 16; pr += 256) {
        const int j = pr & 15, kr = pr >> 4;
        __builtin_prefetch(
            imgn + (size_t)(kb + KC_ + kr) * (T1_ * HW_) + j * HW_ + hw0, 0, 1);
      }
    }
    const float* bbuf = smB[0];
#endif
    // --- Vv GEMM: M=16 channels, N=16 v, K=KC_ per chunk, 16 j tiles ---
#pragma unroll
    for (int ks = 0; ks < KC_ / 4; ++ks) {
      const v2f a = *(const v2f*)(wrow + kb + ks * 4);   // w_iv[c, k..k+1]
#pragma unroll
      for (int j = 0; j < 16; ++j) {
        const v2f b =
            *(const v2f*)(&bbuf[((ks * 16 + j) * 16 + lc) * 4 + 2 * half]);
        acc[j] = wmma_f32(a, b, acc[j]);
      }
    }
#if USE_ASYNC_LDS
    wait_async0();
    __syncthreads();
#endif
  }

  // --- finale: per-channel  y_c(16x16) = att_c(16x16) @ (Vv_c + b_iv[c]) ---
  float* vvw = &smVV[wave * 256];                         // wave-private [v][j]
  const float* attc0 = att_ws + (size_t)(n * C_ + c0) * 256;
  float* outc0 = out + (size_t)(n * C_ + c0) * T1_ * HW_ + hw0;

#pragma unroll 1
  for (int cc = 0; cc < 16; ++cc) {
    const int r = cc & 7;
    if (half == (cc >> 3)) {                // lanes holding channel cc dump it
#pragma unroll
      for (int j = 0; j < 16; ++j) vvw[lc * 16 + j] = acc[j][r];
    }
#if defined(__gfx1250__)
    asm volatile("s_wait_dscnt 0" ::: "memory");   // intra-wave LDS handoff
#endif
    const float bias = b_iv[c0 + cc];
    const float* attc = attc0 + cc * 256;
    v8f y = zero8();
#pragma unroll
    for (int k4 = 0; k4 < 4; ++k4) {
      v2f a = *(const v2f*)(attc + lc * 16 + k4 * 4 + 2 * half);  // att[c,i,j..]
      v2f b = *(const v2f*)(vvw + lc * 16 + k4 * 4 + 2 * half);   // Vv[c,j..,v]
      b.x += bias;
      b.y += bias;
      y = wmma_f32(a, b, y);
    }
    float* oc = outc0 + (size_t)cc * T1_ * HW_;
#pragma unroll
    for (int rr = 0; rr < 8; ++rr) oc[(rr + 8 * half) * HW_ + lc] = y[rr];
  }
}

// ---------------------------------------------------------------------------
extern "C" void kernel_launch(void* const* d_in, const int* in_sizes, int n_in,
                              void* d_out, int out_size, void* d_ws,
                              size_t ws_size, hipStream_t stream) {
  (void)in_sizes; (void)n_in; (void)out_size; (void)ws_size;
  const float* image  = (const float*)d_in[0];
  const float* motion = (const float*)d_in[1];
  const float* w_mq   = (const float*)d_in[2];
  const float* b_mq   = (const float*)d_in[3];
  const float* w_iq   = (const float*)d_in[4];
  const float* b_iq   = (const float*)d_in[5];
  const float* w_iv   = (const float*)d_in[6];
  const float* b_iv   = (const float*)d_in[7];
  const float* w_att  = (const float*)d_in[8];
  const float* b_att  = (const float*)d_in[9];
  float* out = (float*)d_out;

  float* mean_ws = (float*)d_ws;                 // [N][C][T1]  = 32768 floats
  float* att_ws  = mean_ws + N_ * C_ * T1_;      // [N][C][16][16] = 524288 floats

  k_mean<<<N_ * C_ * T1_, 128, 0, stream>>>(image, mean_ws);
  k_att<<<N_, 256, 0, stream>>>(motion, mean_ws, w_mq, b_mq, w_iq, b_iq,
                                w_att, b_att, att_ws);
  k_main<<<dim3(49, 2, 8), 256, 0, stream>>>(image, w_iv, b_iv, att_ws, out);
}